// ClassificationHead_11811160064222
// MI455X (gfx1250) — compile-verified
//
#include <hip/hip_runtime.h>

typedef float v2f __attribute__((ext_vector_type(2)));
typedef float v8f __attribute__((ext_vector_type(8)));

#define Dd    1024
#define NS    25
#define NQ    75
#define NW    5
#define SROW  1028      // padded S row stride in floats (4112B, 16B aligned, conflict-free frags)
#define KROW  32        // augmented [K|Y] row stride
#define WROW  8         // W row stride (d x 5 padded to 8)
#define QROW  132       // Q staging row stride (528B, 16B aligned)
#define KCH   128       // k-chunk for logits GEMM
#define NCHUNK (Dd / KCH)
#define NTHR  160       // 5 waves

// Manual LDS layout (floats):
//   S:    [0, 25700)                 25 x 1028   (reused as per-wave Q staging in stage 5)
//   W:    [25700, 33908)             1024 x 8 + 16 tail pad (zeroed; absorbs phantom tile reads)
//   Kaug: [33908, 34708)             25 x 32  ([K | Y])
//   prow: [34708, 34740)
//   fcol: [34740, 34772)
#define OFF_W    25700
#define OFF_KAUG 33908
#define OFF_PROW 34708
#define OFF_FCOL 34740
#define SMEM_FLOATS 34772

__global__ __launch_bounds__(NTHR) void ridge_head_fused(
    const float* __restrict__ q, const float* __restrict__ s,
    const int* __restrict__ labels, const float* __restrict__ scale,
    float* __restrict__ out)
{
  __shared__ __align__(16) float smem[SMEM_FLOATS];
  float* S_lds = smem;
  float* W_lds = smem + OFF_W;
  float* Kaug  = smem + OFF_KAUG;
  float* prow  = smem + OFF_PROW;
  float* fcol  = smem + OFF_FCOL;

  const int b    = blockIdx.x;
  const int tid  = threadIdx.x;
  const int lane = tid & 31;
  const int wave = tid >> 5;
  const int lm   = lane & 15;   // fragment row/col index
  const int kh   = lane >> 4;   // fragment k-half select

  // ---------------- Stage 1: S -> LDS (coalesced b128), zero W region ----------------
  {
    const float* sb = s + (size_t)b * NS * Dd;
    for (int idx = tid; idx < NS * (Dd / 4); idx += NTHR) {
      int r  = idx >> 8;        // /256 float4 per row
      int c4 = idx & 255;
      float4 v = ((const float4*)(sb + r * Dd))[c4];
      ((float4*)(S_lds + r * SROW))[c4] = v;
    }
    for (int idx = tid; idx < Dd * WROW + 16; idx += NTHR) W_lds[idx] = 0.0f;
  }
  __syncthreads();

  // ---------------- Stage 2: K = S*S^T + I via v_wmma_f32_16x16x4_f32 ----------------
  if (wave < 4) {
    const int ti = (wave >> 1) * 16;       // output tile row base
    const int tj = (wave & 1) * 16;        // output tile col base
    const float* arow = S_lds + (ti + lm) * SROW + 2 * kh;
    const float* brow = S_lds + (tj + lm) * SROW + 2 * kh;
    v8f acc0 = {0.f,0.f,0.f,0.f,0.f,0.f,0.f,0.f};
    v8f acc1 = {0.f,0.f,0.f,0.f,0.f,0.f,0.f,0.f};
    for (int k = 0; k < Dd; k += 8) {
      v2f a0 = *(const v2f*)(arow + k);
      v2f b0 = *(const v2f*)(brow + k);
      acc0 = __builtin_amdgcn_wmma_f32_16x16x4_f32(false, a0, false, b0,
                                                   (short)0, acc0, false, false);
      v2f a1 = *(const v2f*)(arow + k + 4);
      v2f b1 = *(const v2f*)(brow + k + 4);
      acc1 = __builtin_amdgcn_wmma_f32_16x16x4_f32(false, a1, false, b1,
                                                   (short)0, acc1, false, false);
    }
    v8f acc = acc0 + acc1;
    const int j = tj + lm;
    if (j < NS) {
      #pragma unroll
      for (int v = 0; v < 8; ++v) {
        int i = ti + kh * 8 + v;
        if (i < NS) Kaug[i * KROW + j] = acc[v] + ((i == j) ? 1.0f : 0.0f);
      }
    }
  } else {
    // wave 4: one-hot support labels -> Kaug[:, 25..29]
    int r = tid - 128;
    if (r < NS) {
      int lab = labels[b * NS + r];
      for (int c = 0; c < NW; ++c)
        Kaug[r * KROW + NS + c] = (lab == c) ? 1.0f : 0.0f;
    }
  }
  __syncthreads();

  // ---------------- Stage 3: Gauss-Jordan on [K | Y] (25 x 30), SPD => no pivoting ----
  for (int p = 0; p < NS; ++p) {
    if (tid < NS + NW) {
      float pivinv = 1.0f / Kaug[p * KROW + p];
      prow[tid] = Kaug[p * KROW + tid] * pivinv;
      if (tid < NS) fcol[tid] = Kaug[tid * KROW + p];
    }
    __syncthreads();
    for (int e = tid; e < NS * (NS + NW); e += NTHR) {
      int r = e / (NS + NW);
      int c = e - r * (NS + NW);
      float v = (r == p) ? prow[c] : (Kaug[r * KROW + c] - fcol[r] * prow[c]);
      Kaug[r * KROW + c] = v;
    }
    __syncthreads();
  }
  // Now A = Kaug[:, 25..29] = K^{-1} Y

  // ---- Prologue for stage 5: issue chunk-0 Q loads now so HBM latency overlaps stage 4
  const int m0 = wave * 16;                 // per-wave 16-row M tile (5 tiles cover 75)
  const float* qb = q + (size_t)b * NQ * Dd;
  float4 r[16];
  #pragma unroll
  for (int u = 0; u < 16; ++u) {
    int m = m0 + u;
    float4 v = {0.f, 0.f, 0.f, 0.f};
    if (m < NQ) v = ((const float4*)(qb + (size_t)m * Dd))[lane];
    r[u] = v;
  }

  // ---------------- Stage 4: W = S^T A  (1024 x 5, row stride 8) ----------------
  for (int e = tid; e < Dd * NW; e += NTHR) {
    int n = e >> 10;            // 0..4
    int k = e & (Dd - 1);
    float acc = 0.0f;
    #pragma unroll
    for (int i = 0; i < NS; ++i)
      acc += S_lds[i * SROW + k] * Kaug[i * KROW + NS + n];
    W_lds[k * WROW + n] = acc;
  }
  __syncthreads();   // all stage-4 reads of S done before stage 5 overwrites S region

  // ---------------- Stage 5: logits = Q * W via WMMA, register-pipelined staging -----
  // Per-wave private LDS staging region; DS ops are in-order within a wave, so no
  // block barriers are needed. Global loads for chunk c+1 are issued before the WMMA
  // loop over chunk c to overlap HBM latency with matrix math.
  float* Qs = S_lds + wave * (16 * QROW);

  v8f c0 = {0.f,0.f,0.f,0.f,0.f,0.f,0.f,0.f};
  v8f c1 = {0.f,0.f,0.f,0.f,0.f,0.f,0.f,0.f};

  const float* ap = Qs + lm * QROW + 2 * kh;

  for (int c = 0; c < NCHUNK; ++c) {
    // commit current chunk to LDS (waits only on its own global loads)
    #pragma unroll
    for (int u = 0; u < 16; ++u)
      ((float4*)(Qs + u * QROW))[lane] = r[u];

    // prefetch next chunk from HBM into registers
    if (c + 1 < NCHUNK) {
      #pragma unroll
      for (int u = 0; u < 16; ++u) {
        int m = m0 + u;
        float4 v = {0.f, 0.f, 0.f, 0.f};
        if (m < NQ) v = ((const float4*)(qb + (size_t)m * Dd + (c + 1) * KCH))[lane];
        r[u] = v;
      }
    }

    const int k0 = c * KCH;
    #pragma unroll 4
    for (int kc = 0; kc < KCH; kc += 8) {
      int kb0 = k0 + kc + 2 * kh;
      v2f a0 = *(const v2f*)(ap + kc);
      v2f b0 = { W_lds[kb0 * WROW + lm], W_lds[(kb0 + 1) * WROW + lm] };
      c0 = __builtin_amdgcn_wmma_f32_16x16x4_f32(false, a0, false, b0,
                                                 (short)0, c0, false, false);
      int kb1 = kb0 + 4;
      v2f a1 = *(const v2f*)(ap + kc + 4);
      v2f b1 = { W_lds[kb1 * WROW + lm], W_lds[(kb1 + 1) * WROW + lm] };
      c1 = __builtin_amdgcn_wmma_f32_16x16x4_f32(false, a1, false, b1,
                                                 (short)0, c1, false, false);
    }
  }

  const float sc = *scale;
  v8f cc = c0 + c1;
  if (lm < NW) {
    #pragma unroll
    for (int v = 0; v < 8; ++v) {
      int m = m0 + kh * 8 + v;
      if (m < NQ) out[((size_t)b * NQ + m) * NW + lm] = sc * cc[v];
    }
  }
}

extern "C" void kernel_launch(void* const* d_in, const int* in_sizes, int n_in,
                              void* d_out, int out_size, void* d_ws, size_t ws_size,
                              hipStream_t stream) {
  (void)n_in; (void)out_size; (void)d_ws; (void)ws_size;
  const float* q      = (const float*)d_in[0];
  const float* s      = (const float*)d_in[1];
  const int*   labels = (const int*)d_in[2];
  const float* scale  = (const float*)d_in[3];
  float* out = (float*)d_out;

  const int B = in_sizes[1] / (NS * Dd);   // support is (B, 25, 1024)
  if (B <= 0) return;

  ridge_head_fused<<<dim3(B), dim3(NTHR), 0, stream>>>(q, s, labels, scale, out);
}